// CausalConvolution_64244120814214
// MI455X (gfx1250) — compile-verified
//
#include <hip/hip_runtime.h>

typedef float v2f __attribute__((ext_vector_type(2)));
typedef float v8f __attribute__((ext_vector_type(8)));

// Problem constants (from reference: x (32,4,131072) f32, h (32,4,32768) f32)
constexpr int XLEN   = 131072;
constexpr int HLEN   = 32768;
constexpr int NSER   = 128;           // 32*4 independent series
constexpr int SBLK   = 2048;          // output columns per workgroup
constexpr int RCH    = 128;           // r-values (of 2048) staged per LDS chunk
constexpr int NCHUNK = (HLEN / 16) / RCH;     // 16
constexpr int WIN    = SBLK + 16 * RCH;       // 4096 floats, sliding x window
constexpr int HCH    = 16 * RCH;              // 2048 floats of h per chunk
constexpr int TILES  = SBLK / (16 * 8) + 1;   // 17 tiles per wave (1 boundary)

// y[t] = sum_k x[t-k] h[k],  k = 16r + j
//   z[j][s] = sum_r h[16r+j] * x[s - 16r]        (16 x S GEMM, K = 2048)
//   y[t]    = sum_{j=0..15} z[j][t-j]            (diagonal combine, 2 tiles)
__global__ __launch_bounds__(256) void causal_conv_wmma(
    const float* __restrict__ x, const float* __restrict__ h,
    float* __restrict__ y) {
  // Double-buffered staging: async DMA of chunk c+1 overlaps MMA on chunk c.
  __shared__ float xs[2][WIN];     // 2 x 16 KB (buffer 0 reused as z scratch)
  __shared__ float hs[2][HCH];     // 2 x  8 KB

  const int tid  = threadIdx.x;
  const int wave = tid >> 5;
  const int lane = tid & 31;
  const int hh   = lane >> 4;      // lane half (selects K pair / row half)
  const int ln   = lane & 15;      // M for A, N for B/C/D

  const int series = blockIdx.y;
  const int S0     = blockIdx.x * SBLK;
  const float* xp = x + (size_t)series * XLEN;
  const float* hp = h + (size_t)series * HLEN;
  float*       yp = y + (size_t)series * XLEN;

  v8f acc[TILES];
  const v8f vzero = {0.f, 0.f, 0.f, 0.f, 0.f, 0.f, 0.f, 0.f};
#pragma unroll
  for (int i = 0; i < TILES; ++i) acc[i] = vzero;

  // ---- async stage of chunk c into buffer buf (memory -> LDS, ASYNCcnt) ----
  auto stage = [&](int c, int buf) {
    const unsigned xsb = (unsigned)(unsigned long long)&xs[buf][0];
    const unsigned hsb = (unsigned)(unsigned long long)&hs[buf][0];
    const int lo = S0 - 16 * RCH * (c + 1);   // xs[p] = x[lo+p], zero-padded
#pragma unroll
    for (int q = tid; q < WIN; q += 256) {
      const int g = lo + q;
      if ((unsigned)g < (unsigned)XLEN) {
        const unsigned lds  = xsb + 4u * (unsigned)q;
        const unsigned goff = 4u * (unsigned)g;
        asm volatile("global_load_async_to_lds_b32 %0, %1, %2"
                     :: "v"(lds), "v"(goff), "s"(xp) : "memory");
      } else {
        xs[buf][q] = 0.f;  // causal zero pad (DScnt path)
      }
    }
    const int hb = HCH * c;                   // hs[q] = h[16*RCH*c + q]
#pragma unroll
    for (int q = tid; q < HCH; q += 256) {
      const unsigned lds  = hsb + 4u * (unsigned)q;
      const unsigned goff = 4u * (unsigned)(hb + q);
      asm volatile("global_load_async_to_lds_b32 %0, %1, %2"
                   :: "v"(lds), "v"(goff), "s"(hp) : "memory");
    }
  };

  stage(0, 0);
  asm volatile("s_wait_asynccnt 0" ::: "memory");
  __syncthreads();

  for (int c = 0; c < NCHUNK; ++c) {
    const int buf = c & 1;
    if (c + 1 < NCHUNK) stage(c + 1, buf ^ 1);  // DMA next chunk, no wait yet

    const float* xsc = &xs[buf][0];
    const float* hsc = &hs[buf][0];
    // ---- MMA sweep: rr0-outer so A is loaded once per K-step ----
    for (int rr0 = 0; rr0 < RCH; rr0 += 4) {
      // A 16x4 f32: lane holds M=ln, K = {2*hh, 2*hh+1}: A[j][rr]=hs[16*(rr0+rr)+j]
      const int ab = 16 * (rr0 + 2 * hh) + ln;
      v2f a;
      a.x = hsc[ab];
      a.y = hsc[ab + 16];
      // B sliding window: bb(i) = 256*wave + 16*(i-1) + ln + 16*(RCH-rr0-2*hh)
      //   b.x(i)=xs[bb(i)], b.y(i)=xs[bb(i)-16] => b.y(i+1)==b.x(i):
      //   18 loads feed 17 tiles.
      const int base = 256 * wave + ln + 16 * (RCH - rr0 - 2 * hh) - 32;
      float bv[TILES + 1];
#pragma unroll
      for (int j = 0; j <= TILES; ++j) bv[j] = xsc[base + 16 * j];
#pragma unroll
      for (int i = 0; i < TILES; ++i) {
        v2f b;
        b.x = bv[i + 1];
        b.y = bv[i];
        acc[i] = __builtin_amdgcn_wmma_f32_16x16x4_f32(
            /*neg_a=*/false, a, /*neg_b=*/false, b,
            /*c_mod=*/(short)0, acc[i], /*reuse_a=*/false, /*reuse_b=*/false);
      }
    }

    if (c + 1 < NCHUNK) {
      asm volatile("s_wait_asynccnt 0" ::: "memory");  // next chunk landed
      __syncthreads();                                 // all waves swap buffers
    }
  }

  // ---- Diagonal combine. Last chunk computed from buffer 1; buffer 0 was
  // retired at the final inter-chunk barrier, so its xs region is free to use
  // as per-wave z scratch with no further synchronization.
  float* zb = &xs[0][wave * 512];   // zb[slot][j][n], 2*256 floats per wave
  // D layout: VGPR r holds z[j = r + 8*hh][n = ln]
#pragma unroll
  for (int r = 0; r < 8; ++r) zb[0 * 256 + (r + 8 * hh) * 16 + ln] = acc[0][r];

#pragma unroll
  for (int i = 1; i < TILES; ++i) {
    const int slot = i & 1;
#pragma unroll
    for (int r = 0; r < 8; ++r)
      zb[slot * 256 + (r + 8 * hh) * 16 + ln] = acc[i][r];
    asm volatile("s_wait_dscnt 0" ::: "memory");  // z tile visible to wave

    // y[t0 + m] = sum_j z[j][t0 + m - j] ; j<=m this tile, j>m previous tile
    const int m = ln;
    float sum = 0.f;
#pragma unroll
    for (int j = 0; j < 16; ++j) {
      const int sl = (j <= m) ? slot : (slot ^ 1);
      const int n  = (m - j) & 15;
      sum += zb[sl * 256 + j * 16 + n];
    }
    if (hh == 0) {
      const int t0 = S0 + 256 * wave + 16 * (i - 1);
      yp[t0 + m] = sum;
    }
    asm volatile("s_wait_dscnt 0" ::: "memory");  // reads done before reuse
  }
}

extern "C" void kernel_launch(void* const* d_in, const int* in_sizes, int n_in,
                              void* d_out, int out_size, void* d_ws,
                              size_t ws_size, hipStream_t stream) {
  (void)in_sizes; (void)n_in; (void)d_ws; (void)ws_size; (void)out_size;
  const float* x = (const float*)d_in[0];
  const float* h = (const float*)d_in[1];
  float* y = (float*)d_out;
  dim3 grid(XLEN / SBLK, NSER);   // 64 x 128 workgroups
  causal_conv_wmma<<<grid, 256, 0, stream>>>(x, h, y);
}